// LinearAttention_17944373362827
// MI455X (gfx1250) — compile-verified
//
#include <hip/hip_runtime.h>
#include <math.h>

typedef float v2f __attribute__((ext_vector_type(2)));
typedef float v4f __attribute__((ext_vector_type(4)));
typedef float v8f __attribute__((ext_vector_type(8)));

#define BB 8
#define SS 4096
#define EE 768
#define HH 12
#define DD 64
#define KDIM 768              // K of both big GEMMs (E == 768)
#define MDIM (BB * SS)        // 32768 rows

__device__ __forceinline__ v8f wmma_f32(v2f a, v2f b, v8f c) {
  // D = A(16x4 f32) * B(4x16 f32) + C(16x16 f32)
  return __builtin_amdgcn_wmma_f32_16x16x4_f32(
      /*neg_a=*/false, a, /*neg_b=*/false, b,
      /*c_mod=*/(short)0, c, /*reuse_a=*/false, /*reuse_b=*/false);
}

__device__ __forceinline__ float phi(float x) {  // elu(x) + 1
  return x > 0.f ? x + 1.f : expf(x);
}

// ---------------------------------------------------------------------------
// Tiled 128x128 fp32 WMMA GEMM, double-buffered LDS with register staging.
// MODE 0: QKV epilogue (feature map + [B,H,S,D] scatter).
// MODE 1: projection epilogue (+bias, row-major [B,S,E] out).
// ---------------------------------------------------------------------------
template <int MODE>
__global__ __launch_bounds__(256) void gemm128(
    const float* __restrict__ A, const float* __restrict__ W, int N,
    float* __restrict__ O0, float* __restrict__ O1, float* __restrict__ O2,
    const float* __restrict__ bias) {
  __shared__ float As[2][128 * 20];   // lda = 20 (padded)
  __shared__ float Bs[2][16 * 136];   // ldb = 136 (padded)

  const int tid = threadIdx.x;
  const int lane = tid & 31;
  const int wid = tid >> 5;
  const int l16 = lane & 15;
  const int hs = lane >> 4;                 // half-wave select: K offset 0/2
  const int m0 = blockIdx.y * 128;
  const int n0 = blockIdx.x * 128;
  const int wm = (wid >> 2) * 64;           // wave row base (0 / 64)
  const int wn = (wid & 3) * 32;            // wave col base (0..96)

  const v8f vzero = {0.f, 0.f, 0.f, 0.f, 0.f, 0.f, 0.f, 0.f};
  v8f acc[4][2];
#pragma unroll
  for (int i = 0; i < 4; ++i)
#pragma unroll
    for (int j = 0; j < 2; ++j) acc[i][j] = vzero;

  const int arow = tid >> 2;                // 0..63
  const int acol = (tid & 3) << 2;          // 0,4,8,12
  const int brow = tid >> 5;                // 0..7 (rb0) / +8 (rb1)
  const int bcol = (tid & 31) << 2;         // 0..124

  const float* aptr0 = A + (size_t)(m0 + arow) * KDIM + acol;
  const float* aptr1 = A + (size_t)(m0 + arow + 64) * KDIM + acol;
  const float* bptr0 = W + (size_t)brow * N + n0 + bcol;
  const float* bptr1 = W + (size_t)(brow + 8) * N + n0 + bcol;

  // preload tile k0 = 0 into registers
  v4f ra0 = *(const v4f*)(aptr0);
  v4f ra1 = *(const v4f*)(aptr1);
  v4f rb0 = *(const v4f*)(bptr0);
  v4f rb1 = *(const v4f*)(bptr1);

  int buf = 0;
  for (int k0 = 0; k0 < KDIM; k0 += 16) {
    // stage current tile to LDS
    *(v4f*)&As[buf][arow * 20 + acol] = ra0;
    *(v4f*)&As[buf][(arow + 64) * 20 + acol] = ra1;
    *(v4f*)&Bs[buf][brow * 136 + bcol] = rb0;
    *(v4f*)&Bs[buf][(brow + 8) * 136 + bcol] = rb1;
    __syncthreads();

    // issue next-tile global loads so they overlap with the WMMA block
    if (k0 + 16 < KDIM) {
      ra0 = *(const v4f*)(aptr0 + k0 + 16);
      ra1 = *(const v4f*)(aptr1 + k0 + 16);
      rb0 = *(const v4f*)(bptr0 + (size_t)(k0 + 16) * N);
      rb1 = *(const v4f*)(bptr1 + (size_t)(k0 + 16) * N);
      if (k0 + 32 < KDIM) {
        __builtin_prefetch(aptr0 + k0 + 32, 0, 3);
        __builtin_prefetch(aptr1 + k0 + 32, 0, 3);
        __builtin_prefetch(bptr0 + (size_t)(k0 + 32) * N, 0, 3);
        __builtin_prefetch(bptr1 + (size_t)(k0 + 32) * N, 0, 3);
      }
    }

#pragma unroll
    for (int kk = 0; kk < 16; kk += 4) {
      const int ka = kk + hs * 2;
      v2f a[4], b[2];
#pragma unroll
      for (int i = 0; i < 4; ++i)
        a[i] = *(const v2f*)&As[buf][(wm + i * 16 + l16) * 20 + ka];
#pragma unroll
      for (int j = 0; j < 2; ++j) {
        b[j].x = Bs[buf][ka * 136 + wn + j * 16 + l16];
        b[j].y = Bs[buf][(ka + 1) * 136 + wn + j * 16 + l16];
      }
#pragma unroll
      for (int i = 0; i < 4; ++i)
#pragma unroll
        for (int j = 0; j < 2; ++j) acc[i][j] = wmma_f32(a[i], b[j], acc[i][j]);
    }
    buf ^= 1;
  }

#pragma unroll
  for (int i = 0; i < 4; ++i) {
#pragma unroll
    for (int j = 0; j < 2; ++j) {
      const int n = n0 + wn + j * 16 + l16;
#pragma unroll
      for (int p = 0; p < 8; ++p) {
        const int m = m0 + wm + i * 16 + p + hs * 8;
        const float val = acc[i][j][p];
        if (MODE == 0) {
          const int t = n / (HH * DD);          // 0=q 1=k 2=v
          const int rem = n - t * (HH * DD);
          const int h = rem >> 6;
          const int d = rem & 63;
          const int b_ = m >> 12;               // / 4096
          const int s = m & 4095;
          const size_t off = ((size_t)(b_ * HH + h) * SS + s) * DD + d;
          if (t == 0)      O0[off] = phi(val);
          else if (t == 1) O1[off] = phi(val);
          else             O2[off] = val;
        } else {
          O0[(size_t)m * EE + n] = val + bias[n];
        }
      }
    }
  }
}

// ---------------------------------------------------------------------------
// kv[b,h,d,e] = sum_s phi(k)[b,h,s,d] * v[b,h,s,e]   (64x64 per head)
// One block per (b,h); 8 waves split S; LDS ds_add reduction.
// ---------------------------------------------------------------------------
__global__ __launch_bounds__(256) void kv_kernel(
    const float* __restrict__ Kf, const float* __restrict__ Vf,
    float* __restrict__ KV) {
  __shared__ float kvs[64 * 64];
  const int tid = threadIdx.x, lane = tid & 31, wid = tid >> 5;
  const int l16 = lane & 15, hs = lane >> 4;
  const size_t base = (size_t)blockIdx.x * SS * DD;
  const float* kp = Kf + base;
  const float* vp = Vf + base;

  for (int i = tid; i < 4096; i += 256) kvs[i] = 0.f;

  const v8f vzero = {0.f, 0.f, 0.f, 0.f, 0.f, 0.f, 0.f, 0.f};
  v8f acc[4][4];
#pragma unroll
  for (int i = 0; i < 4; ++i)
#pragma unroll
    for (int j = 0; j < 4; ++j) acc[i][j] = vzero;

  const int sBeg = wid * (SS / 8);
  for (int s0 = sBeg; s0 < sBeg + SS / 8; s0 += 4) {
    const int sa = s0 + hs * 2;
    v2f a[4], b[4];
#pragma unroll
    for (int i = 0; i < 4; ++i) {
      a[i].x = kp[(size_t)sa * DD + i * 16 + l16];        // A = k^T (M=d,K=s)
      a[i].y = kp[(size_t)(sa + 1) * DD + i * 16 + l16];
      b[i].x = vp[(size_t)sa * DD + i * 16 + l16];        // B = v   (K=s,N=e)
      b[i].y = vp[(size_t)(sa + 1) * DD + i * 16 + l16];
    }
#pragma unroll
    for (int i = 0; i < 4; ++i)
#pragma unroll
      for (int j = 0; j < 4; ++j) acc[i][j] = wmma_f32(a[i], b[j], acc[i][j]);
  }
  __syncthreads();
#pragma unroll
  for (int i = 0; i < 4; ++i)
#pragma unroll
    for (int j = 0; j < 4; ++j)
#pragma unroll
      for (int p = 0; p < 8; ++p)
        atomicAdd(&kvs[(i * 16 + p + hs * 8) * 64 + j * 16 + l16],
                  acc[i][j][p]);
  __syncthreads();
  float* dst = KV + (size_t)blockIdx.x * 4096;
  for (int i = tid; i < 4096; i += 256) dst[i] = kvs[i];
}

// k_sum[b,h,d] = sum_s phi(k)[b,h,s,d]
__global__ __launch_bounds__(256) void ksum_kernel(
    const float* __restrict__ Kf, float* __restrict__ Ks) {
  __shared__ float part[4][64];
  const float* kp = Kf + (size_t)blockIdx.x * SS * DD;
  const int d = threadIdx.x & 63, q = threadIdx.x >> 6;
  float acc = 0.f;
  for (int s = q * 1024; s < q * 1024 + 1024; ++s)
    acc += kp[(size_t)s * DD + d];
  part[q][d] = acc;
  __syncthreads();
  if (threadIdx.x < 64)
    Ks[blockIdx.x * 64 + d] = part[0][d] + part[1][d] + part[2][d] + part[3][d];
}

// out[b,h,s,e] = (q[b,h,s,:] @ kv[b,h]) / (q[b,h,s,:] . k_sum[b,h])
// One block per (b, h, 128-row s-block); result written in [B,S,E] layout.
__global__ __launch_bounds__(256) void attn_kernel(
    const float* __restrict__ Qf, const float* __restrict__ KV,
    const float* __restrict__ Ks, float* __restrict__ Attn) {
  __shared__ float kvs[64 * 72];     // ldb = 72 (padded)
  __shared__ float ks[64];
  __shared__ float qs[8][16 * 68];   // per-wave q tile, lda = 68 (padded)
  __shared__ float zs[8][16];

  const int tid = threadIdx.x, lane = tid & 31, wid = tid >> 5;
  const int l16 = lane & 15, hs = lane >> 4;
  const int blk = blockIdx.x;
  const int sb = blk & 31;           // S/128 = 32 blocks per head
  const int bh = blk >> 5;
  const int h = bh % HH, b = bh / HH;

  for (int i = tid; i < 4096; i += 256)
    kvs[(i >> 6) * 72 + (i & 63)] = KV[(size_t)bh * 4096 + i];
  if (tid < 64) ks[tid] = Ks[bh * 64 + tid];

  const int sBase = sb * 128 + wid * 16;
  const float* qp = Qf + ((size_t)bh * SS + sBase) * DD;
#pragma unroll
  for (int it = 0; it < 8; ++it) {
    const int flat = it * 32 + lane;
    const int row = flat >> 4;
    const int c4 = (flat & 15) << 2;
    *(v4f*)&qs[wid][row * 68 + c4] = *(const v4f*)&qp[(size_t)row * DD + c4];
  }
  __syncthreads();

  if (lane < 16) {                   // z = 1 / (q . k_sum) per row
    float dot = 0.f;
    for (int d = 0; d < 64; ++d) dot += qs[wid][lane * 68 + d] * ks[d];
    zs[wid][lane] = 1.f / dot;
  }
  __syncthreads();

  const v8f vzero = {0.f, 0.f, 0.f, 0.f, 0.f, 0.f, 0.f, 0.f};
#pragma unroll
  for (int j = 0; j < 4; ++j) {
    v8f acc = vzero;
#pragma unroll
    for (int d0 = 0; d0 < 64; d0 += 4) {
      const int ka = d0 + hs * 2;
      v2f a = *(const v2f*)&qs[wid][l16 * 68 + ka];
      v2f bb;
      bb.x = kvs[ka * 72 + j * 16 + l16];
      bb.y = kvs[(ka + 1) * 72 + j * 16 + l16];
      acc = wmma_f32(a, bb, acc);
    }
#pragma unroll
    for (int p = 0; p < 8; ++p) {
      const int row = p + hs * 8;
      const int s = sBase + row;
      const float val = acc[p] * zs[wid][row];
      Attn[((size_t)b * SS + s) * EE + h * 64 + j * 16 + l16] = val;
    }
  }
}

extern "C" void kernel_launch(void* const* d_in, const int* in_sizes, int n_in,
                              void* d_out, int out_size, void* d_ws,
                              size_t ws_size, hipStream_t stream) {
  (void)in_sizes; (void)n_in; (void)out_size; (void)ws_size;
  const float* x      = (const float*)d_in[0];
  const float* w_qkv  = (const float*)d_in[1];
  const float* w_proj = (const float*)d_in[2];
  const float* b_proj = (const float*)d_in[3];

  float* out = (float*)d_out;                       // [B,S,E] flat
  float* kv_out = out + (size_t)BB * SS * EE;       // [B,H,64,64] flat

  const size_t SZ = (size_t)BB * HH * SS * DD;      // 25,165,824 floats
  float* ws   = (float*)d_ws;
  float* Qf   = ws;            // phi(q)  [B,H,S,D]
  float* Kf   = ws + SZ;       // phi(k)  [B,H,S,D]
  float* Vf   = ws + 2 * SZ;   // v       [B,H,S,D]
  float* Attn = ws + 3 * SZ;   // pre-proj output [B,S,E]
  float* Ksum = ws + 4 * SZ;   // [B,H,D]

  // 1) QKV GEMM + feature map + layout transform
  gemm128<0><<<dim3(3 * EE / 128, MDIM / 128), 256, 0, stream>>>(
      x, w_qkv, 3 * EE, Qf, Kf, Vf, nullptr);
  // 2) kv = k^T v per head (written straight into d_out's kv region)
  kv_kernel<<<dim3(BB * HH), 256, 0, stream>>>(Kf, Vf, kv_out);
  // 3) k_sum
  ksum_kernel<<<dim3(BB * HH), 256, 0, stream>>>(Kf, Ksum);
  // 4) out = (q @ kv) * 1/(q . k_sum), to [B,S,E]
  attn_kernel<<<dim3(BB * HH * (SS / 128)), 256, 0, stream>>>(
      Qf, kv_out, Ksum, Attn);
  // 5) projection + bias
  gemm128<1><<<dim3(EE / 128, MDIM / 128), 256, 0, stream>>>(
      Attn, w_proj, EE, out, nullptr, nullptr, b_proj);
}